// GCN_74869869904461
// MI455X (gfx1250) — compile-verified
//
#include <hip/hip_runtime.h>
#include <hip/hip_bf16.h>
#include <math.h>

typedef __attribute__((ext_vector_type(2))) float v2f;
typedef __attribute__((ext_vector_type(8))) float v8f;

// ---------------------------------------------------------------------------
// Degree / normalization (+ int64 -> int32 index narrowing)
// ---------------------------------------------------------------------------
__global__ void k_init_deg(float* deg, int n) {
    int i = blockIdx.x * blockDim.x + threadIdx.x;
    if (i < n) deg[i] = 1.0f;  // self-loop contributes 1 to in-degree
}

// Fallback (workspace too small for index cache): degree only, int64 reads
__global__ void k_deg_edges(const long long* __restrict__ dst, float* deg, int E) {
    int e = blockIdx.x * blockDim.x + threadIdx.x;
    if (e < E) atomicAdd(&deg[(int)dst[e]], 1.0f);
}

// Preferred: degree + narrow both index rows to int32 in one pass
__global__ void k_prep_edges(const long long* __restrict__ src64,
                             const long long* __restrict__ dst64,
                             int* __restrict__ src32, int* __restrict__ dst32,
                             float* deg, int E) {
    int e = blockIdx.x * blockDim.x + threadIdx.x;
    if (e >= E) return;
    int s = (int)src64[e];
    int d = (int)dst64[e];
    src32[e] = s;
    dst32[e] = d;
    atomicAdd(&deg[d], 1.0f);
}

__global__ void k_dinv(float* deg, int n) {
    int i = blockIdx.x * blockDim.x + threadIdx.x;
    if (i < n) {
        float d = deg[i];
        deg[i] = (d > 0.0f) ? rsqrtf(d) : 0.0f;  // in-place: deg -> dinv
    }
}

// ---------------------------------------------------------------------------
// x @ W1 via V_WMMA_F32_16X16X4_F32 : A = 16 nodes x K4 (x padded 3->4),
// B = W1 padded to 4x16, D = 16x16 f32 tile.
// ---------------------------------------------------------------------------
__global__ void k_xw1_wmma(const float* __restrict__ x, const float* __restrict__ W1,
                           float* __restrict__ h, int n_tiles) {
    int wave = (blockIdx.x * blockDim.x + threadIdx.x) >> 5;
    int lane = threadIdx.x & 31;
    if (wave >= n_tiles) return;            // wave-uniform: EXEC stays all-ones
    int half = lane >> 4;                   // 0: lanes 0-15, 1: lanes 16-31
    int m    = lane & 15;

    // A-matrix (16x4, MxK): lanes 0-15 hold K={0,1}; lanes 16-31 hold K={2,3}
    int node = wave * 16 + m;
    v2f a;
    if (half == 0) { a.x = x[node * 3 + 0]; a.y = x[node * 3 + 1]; }
    else           { a.x = x[node * 3 + 2]; a.y = 0.0f; }

    // B-matrix (4x16, KxN): VGPR0 = row (0 | 2), VGPR1 = row (1 | 3) per lane-half
    v2f b;
    if (half == 0) { b.x = W1[0 * 16 + m]; b.y = W1[1 * 16 + m]; }
    else           { b.x = W1[2 * 16 + m]; b.y = 0.0f; }

    v8f c = {};
    c = __builtin_amdgcn_wmma_f32_16x16x4_f32(false, a, false, b, (short)0, c, false, false);

    // D (16x16): VGPR r -> row = half*8 + r, col = m
#pragma unroll
    for (int r = 0; r < 8; ++r) {
        int row = wave * 16 + half * 8 + r;
        h[(size_t)row * 16 + m] = c[r];
    }
}

// Scalar tail for remainder nodes (N=100000 divisible by 16 -> unused, kept safe)
__global__ void k_xw1_tail(const float* __restrict__ x, const float* __restrict__ W1,
                           float* __restrict__ h, int start, int n) {
    int node = start + blockIdx.x * blockDim.x + threadIdx.x;
    if (node >= n) return;
    float x0 = x[node * 3 + 0], x1 = x[node * 3 + 1], x2 = x[node * 3 + 2];
#pragma unroll
    for (int f = 0; f < 16; ++f)
        h[(size_t)node * 16 + f] = x0 * W1[f] + x1 * W1[16 + f] + x2 * W1[32 + f];
}

// ---------------------------------------------------------------------------
// Layer-1 propagation
// ---------------------------------------------------------------------------
// agg1 = dinv^2 * h  (self-loop term), float4-vectorized sweep
__global__ void k_selfloop1(const float* __restrict__ dinv, const float* __restrict__ h,
                            float* __restrict__ agg1, int n) {
    int i = blockIdx.x * blockDim.x + threadIdx.x;  // group of 4 floats
    int node = i >> 2;                               // 4 groups per node
    if (node >= n) return;
    float di = dinv[node];
    float s = di * di;
    float4 v = reinterpret_cast<const float4*>(h)[i];
    v.x *= s; v.y *= s; v.z *= s; v.w *= s;
    reinterpret_cast<float4*>(agg1)[i] = v;
}

template <typename IT>
__global__ void k_edge1(const IT* __restrict__ src, const IT* __restrict__ dst,
                        const float* __restrict__ dinv, const float* __restrict__ h,
                        float* __restrict__ agg1, int E) {
    int e = blockIdx.x * blockDim.x + threadIdx.x;
    if (e >= E) return;
    int s = (int)src[e];
    int d = (int)dst[e];
    float norm = dinv[s] * dinv[d];
    const float4* hs = reinterpret_cast<const float4*>(h + (size_t)s * 16);
    float4 h0 = hs[0], h1 = hs[1], h2 = hs[2], h3 = hs[3];  // 2x global_load_b128 pairs
    float* ad = agg1 + (size_t)d * 16;
    atomicAdd(ad + 0,  norm * h0.x); atomicAdd(ad + 1,  norm * h0.y);
    atomicAdd(ad + 2,  norm * h0.z); atomicAdd(ad + 3,  norm * h0.w);
    atomicAdd(ad + 4,  norm * h1.x); atomicAdd(ad + 5,  norm * h1.y);
    atomicAdd(ad + 6,  norm * h1.z); atomicAdd(ad + 7,  norm * h1.w);
    atomicAdd(ad + 8,  norm * h2.x); atomicAdd(ad + 9,  norm * h2.y);
    atomicAdd(ad + 10, norm * h2.z); atomicAdd(ad + 11, norm * h2.w);
    atomicAdd(ad + 12, norm * h3.x); atomicAdd(ad + 13, norm * h3.y);
    atomicAdd(ad + 14, norm * h3.z); atomicAdd(ad + 15, norm * h3.w);
}

// ---------------------------------------------------------------------------
// JAX threefry-2x32 dropout (key = (0, 42), p = 0.5)
// keep  <=>  top bit of generated 32-bit word is 0
// ---------------------------------------------------------------------------
__device__ __forceinline__ unsigned rotl32(unsigned x, int r) {
    return (x << r) | (x >> (32 - r));
}

__device__ bool dropout_keep(unsigned i, unsigned half_total) {
    unsigned x0, x1;
    bool first;
    if (i < half_total) { x0 = i; x1 = i + half_total; first = true; }
    else                { x0 = i - half_total; x1 = i; first = false; }
    const unsigned ks0 = 0u, ks1 = 42u;
    const unsigned ks2 = 0x1BD11BDAu ^ ks0 ^ ks1;
    x0 += ks0; x1 += ks1;
    const int rA[4] = {13, 15, 26, 6};
    const int rB[4] = {17, 29, 16, 24};
#define TF_FOUR(RR) _Pragma("unroll") for (int j = 0; j < 4; ++j) { x0 += x1; x1 = rotl32(x1, RR[j]); x1 ^= x0; }
    TF_FOUR(rA); x0 += ks1; x1 += ks2 + 1u;
    TF_FOUR(rB); x0 += ks2; x1 += ks0 + 2u;
    TF_FOUR(rA); x0 += ks0; x1 += ks1 + 3u;
    TF_FOUR(rB); x0 += ks1; x1 += ks2 + 4u;
    TF_FOUR(rA); x0 += ks2; x1 += ks0 + 5u;
#undef TF_FOUR
    unsigned bits = first ? x0 : x1;
    return (bits >> 31) == 0u;  // uniform_[1,2)-1 < 0.5
}

// ---------------------------------------------------------------------------
// Node kernel: ELU + dropout + (h @ W2), self-loop init of agg2
// ---------------------------------------------------------------------------
__global__ void k_node1(const float* __restrict__ agg1, const float* __restrict__ b1,
                        const float* __restrict__ W2, const float* __restrict__ dinv,
                        float* __restrict__ z, float* __restrict__ agg2,
                        int n, unsigned half_total) {
    int i = blockIdx.x * blockDim.x + threadIdx.x;
    if (i >= n) return;
    float acc = 0.0f;
#pragma unroll
    for (int f = 0; f < 16; ++f) {
        float v = agg1[(size_t)i * 16 + f] + b1[f];
        v = (v > 0.0f) ? v : expm1f(v);                    // ELU
        unsigned flat = (unsigned)i * 16u + (unsigned)f;
        v = dropout_keep(flat, half_total) ? v * 2.0f : 0.0f;
        acc += v * W2[f];
    }
    z[i] = acc;
    float di = dinv[i];
    agg2[i] = di * di * acc;  // self-loop contribution, layer 2
}

template <typename IT>
__global__ void k_edge2(const IT* __restrict__ src, const IT* __restrict__ dst,
                        const float* __restrict__ dinv, const float* __restrict__ z,
                        float* __restrict__ agg2, int E) {
    int e = blockIdx.x * blockDim.x + threadIdx.x;
    if (e >= E) return;
    int s = (int)src[e];
    int d = (int)dst[e];
    atomicAdd(&agg2[d], dinv[s] * dinv[d] * z[s]);
}

__global__ void k_final(const float* __restrict__ agg2, const float* __restrict__ b2,
                        float* __restrict__ out, int n) {
    int i = blockIdx.x * blockDim.x + threadIdx.x;
    if (i >= n) return;
    float v = agg2[i] + b2[0];
    out[i] = (v > 0.0f) ? v : expm1f(v);  // final ELU
}

// ---------------------------------------------------------------------------
// Host launcher
// ---------------------------------------------------------------------------
extern "C" void kernel_launch(void* const* d_in, const int* in_sizes, int n_in,
                              void* d_out, int out_size, void* d_ws, size_t ws_size,
                              hipStream_t stream) {
    const float*     x   = (const float*)d_in[0];
    const long long* ei  = (const long long*)d_in[1];
    const float*     W1  = (const float*)d_in[2];
    const float*     b1  = (const float*)d_in[3];
    const float*     W2  = (const float*)d_in[4];
    const float*     b2  = (const float*)d_in[5];
    float*           out = (float*)d_out;

    const int n = in_sizes[0] / 3;       // 100000
    const int E = in_sizes[1] / 2;       // 3200000
    const long long* src = ei;           // edge_index[0]
    const long long* dst = ei + E;       // edge_index[1]

    float* ws   = (float*)d_ws;
    float* dinv = ws;                          // n   (deg, then dinv in place)
    float* h    = dinv + n;                    // n*16 (64B-aligned rows)
    float* agg1 = h + (size_t)n * 16;          // n*16
    float* z    = agg1 + (size_t)n * 16;       // n
    float* agg2 = z + n;                       // n

    const size_t float_bytes = (size_t)(35) * (size_t)n * sizeof(float);
    const bool use_i32 = ws_size >= float_bytes + (size_t)E * 2u * sizeof(int);
    int* src32 = (int*)((char*)d_ws + float_bytes);
    int* dst32 = src32 + E;

    const int B = 256;
    const int gN   = (n + B - 1) / B;
    const int gNF4 = (n * 4 + B - 1) / B;    // float4 sweep over n*16 floats
    const int gE   = (E + B - 1) / B;
    const int tiles = n / 16;
    const int rem   = n - tiles * 16;
    const int wavesPerBlock = B / 32;
    const int gT = (tiles + wavesPerBlock - 1) / wavesPerBlock;

    // 1) degrees (+ optional index narrowing) -> dinv
    k_init_deg<<<gN, B, 0, stream>>>(dinv, n);
    if (use_i32)
        k_prep_edges<<<gE, B, 0, stream>>>(src, dst, src32, dst32, dinv, E);
    else
        k_deg_edges<<<gE, B, 0, stream>>>(dst, dinv, E);
    k_dinv<<<gN, B, 0, stream>>>(dinv, n);

    // 2) h = x @ W1 (WMMA f32 16x16x4)
    k_xw1_wmma<<<gT, B, 0, stream>>>(x, W1, h, tiles);
    if (rem > 0)
        k_xw1_tail<<<1, 32, 0, stream>>>(x, W1, h, tiles * 16, n);

    // 3) layer-1 propagation
    k_selfloop1<<<gNF4, B, 0, stream>>>(dinv, h, agg1, n);
    if (use_i32)
        k_edge1<int><<<gE, B, 0, stream>>>(src32, dst32, dinv, h, agg1, E);
    else
        k_edge1<long long><<<gE, B, 0, stream>>>(src, dst, dinv, h, agg1, E);

    // 4) ELU + dropout + W2 transform + self-loop init for layer 2
    k_node1<<<gN, B, 0, stream>>>(agg1, b1, W2, dinv, z, agg2, n, (unsigned)(n * 8));

    // 5) layer-2 propagation + final ELU
    if (use_i32)
        k_edge2<int><<<gE, B, 0, stream>>>(src32, dst32, dinv, z, agg2, E);
    else
        k_edge2<long long><<<gE, B, 0, stream>>>(src, dst, dinv, z, agg2, E);
    k_final<<<gN, B, 0, stream>>>(agg2, b2, out, n);
}